// IOULoss_28269474742753
// MI455X (gfx1250) — compile-verified
//
#include <hip/hip_runtime.h>

#define MAXV 8
#define EPSF 1e-7f
#define NBLK 2048
#define TPB  256

struct P2 { float x, y; };

__device__ __forceinline__ float edge_cross(P2 A, P2 B, P2 P) {
    return (B.x - A.x) * (P.y - A.y) - (B.y - A.y) * (P.x - A.x);
}

// Branchless conditional dynamic-index write into a register array.
// jmax is a compile-time constant at every call site (i2 is unrolled).
__device__ __forceinline__ void set_at_if(P2* a, int idx, P2 v, bool cond, int jmax) {
#pragma unroll
    for (int j = 0; j < MAXV; ++j) {
        if (j <= jmax) {
            bool w = cond && (j == idx);
            a[j].x = w ? v.x : a[j].x;
            a[j].y = w ? v.y : a[j].y;
        }
    }
}

__device__ float diou_single(const float4* __restrict__ pv,
                             const float4* __restrict__ tv, int i) {
    float4 p01 = pv[2 * i], p23 = pv[2 * i + 1];
    float4 t01 = tv[2 * i], t23 = tv[2 * i + 1];
    P2 p[4] = {{p01.x, p01.y}, {p01.z, p01.w}, {p23.x, p23.y}, {p23.z, p23.w}};
    P2 t[4] = {{t01.x, t01.y}, {t01.z, t01.w}, {t23.x, t23.y}, {t23.z, t23.w}};

    float sa_p = 0.5f * (p[0].x * p[1].y - p[0].y * p[1].x +
                         p[1].x * p[2].y - p[1].y * p[2].x +
                         p[2].x * p[3].y - p[2].y * p[3].x +
                         p[3].x * p[0].y - p[3].y * p[0].x);
    float sa_t = 0.5f * (t[0].x * t[1].y - t[0].y * t[1].x +
                         t[1].x * t[2].y - t[1].y * t[2].x +
                         t[2].x * t[3].y - t[2].y * t[3].x +
                         t[3].x * t[0].y - t[3].y * t[0].x);
    if (sa_p < 0.0f) { P2 tmp = p[0]; p[0] = p[3]; p[3] = tmp; tmp = p[1]; p[1] = p[2]; p[2] = tmp; }
    if (sa_t < 0.0f) { P2 tmp = t[0]; t[0] = t[3]; t[3] = tmp; tmp = t[1]; t[1] = t[2]; t[2] = tmp; }
    float area_p = fabsf(sa_p);
    float area_t = fabsf(sa_t);

    P2 buf[MAXV];
#pragma unroll
    for (int j = 0; j < MAXV; ++j) { buf[j].x = 0.0f; buf[j].y = 0.0f; }
    buf[0] = p[0]; buf[1] = p[1]; buf[2] = p[2]; buf[3] = p[3];
    int n = 4;

    // Provable upper bound on the vertex count entering each clip pass:
    // pass0 input = 4 exactly; one half-plane clip of an n-gon yields
    // <= inside + 2*min(inside, n-inside) <= ceil(3n/2) vertices,
    // so pass1 input <= 6; afterwards bounded by buffer capacity (8).
    const int PASS_BOUND[4] = {4, 6, MAXV, MAXV};

#pragma unroll
    for (int k = 0; k < 4; ++k) {
        P2 A = t[k];
        P2 B = t[(k + 1) & 3];
        // Per-vertex signed distances, computed once with static indexing.
        float d[MAXV];
#pragma unroll
        for (int j = 0; j < MAXV; ++j) d[j] = edge_cross(A, B, buf[j]);

        P2 out[MAXV];
#pragma unroll
        for (int j = 0; j < MAXV; ++j) { out[j].x = 0.0f; out[j].y = 0.0f; }
        int m = 0;
        const int nn = (n > 1) ? n : 1;
        const int L = PASS_BOUND[k];  // compile-time (k unrolled); n <= L always
#pragma unroll
        for (int i2 = 0; i2 < MAXV; ++i2) {
            if (i2 >= L) continue;  // iterations with i2 >= n never write; n <= L
            bool valid = (i2 < n);
            P2 cur = buf[i2];
            float dc = d[i2];
            // JAX semantics: idx = mod(i2+1, nn), then gather CLAMPS idx to 7.
            // For valid iters idx is 0 or i2+1, except the nn>8 overflow case at
            // i2==7 where mod(8,nn)==8 clamps to 7.
            P2 nxt; float dn;
            if (i2 + 1 < MAXV) {
                bool wrap = (i2 + 1 >= nn);
                nxt.x = wrap ? buf[0].x : buf[i2 + 1].x;
                nxt.y = wrap ? buf[0].y : buf[i2 + 1].y;
                dn    = wrap ? d[0]     : d[i2 + 1];
            } else {  // i2 == 7: wrap to 0 iff nn <= 8, else JAX clamp -> slot 7
                bool wrap = (MAXV >= nn);
                nxt.x = wrap ? buf[0].x : buf[MAXV - 1].x;
                nxt.y = wrap ? buf[0].y : buf[MAXV - 1].y;
                dn    = wrap ? d[0]     : d[MAXV - 1];
            }
            float denom = dc - dn;
            float tt = dc / ((fabsf(denom) < EPSF) ? 1.0f : denom);
            P2 ip = { cur.x + tt * (nxt.x - cur.x), cur.y + tt * (nxt.y - cur.y) };

            bool add_cur = valid && (dc >= 0.0f);
            {   // before this write m <= 2*i2
                const int jmax = (2 * i2 < MAXV - 1) ? 2 * i2 : MAXV - 1;
                set_at_if(out, (m < MAXV - 1) ? m : (MAXV - 1), cur, add_cur, jmax);
            }
            m += add_cur ? 1 : 0;

            bool add_ip = valid && ((dc >= 0.0f) != (dn >= 0.0f));
            {   // before this write m <= 2*i2+1
                const int jmax = (2 * i2 + 1 < MAXV - 1) ? 2 * i2 + 1 : MAXV - 1;
                set_at_if(out, (m < MAXV - 1) ? m : (MAXV - 1), ip, add_ip, jmax);
            }
            m += add_ip ? 1 : 0;
        }
#pragma unroll
        for (int j = 0; j < MAXV; ++j) buf[j] = out[j];
        n = m;
    }

    // masked shoelace of the clipped polygon
    float acc = 0.0f;
    {
        const int nn = (n > 1) ? n : 1;
#pragma unroll
        for (int i2 = 0; i2 < MAXV; ++i2) {
            bool valid = (i2 < n);
            P2 cur = buf[i2];
            P2 nxt;
            if (i2 + 1 < MAXV) {
                bool wrap = (i2 + 1 >= nn);
                nxt.x = wrap ? buf[0].x : buf[i2 + 1].x;
                nxt.y = wrap ? buf[0].y : buf[i2 + 1].y;
            } else {
                bool wrap = (MAXV >= nn);
                nxt.x = wrap ? buf[0].x : buf[MAXV - 1].x;
                nxt.y = wrap ? buf[0].y : buf[MAXV - 1].y;
            }
            float term = cur.x * nxt.y - cur.y * nxt.x;
            acc += valid ? term : 0.0f;
        }
    }
    float inter = fabsf(0.5f * acc);
    float uni = area_p + area_t - inter;
    float iou = inter / (uni + EPSF);

    float cpx = 0.25f * (p[0].x + p[1].x + p[2].x + p[3].x);
    float cpy = 0.25f * (p[0].y + p[1].y + p[2].y + p[3].y);
    float ctx = 0.25f * (t[0].x + t[1].x + t[2].x + t[3].x);
    float cty = 0.25f * (t[0].y + t[1].y + t[2].y + t[3].y);
    float d2 = (cpx - ctx) * (cpx - ctx) + (cpy - cty) * (cpy - cty);

    float mxx = p[0].x, mnx = p[0].x, mxy = p[0].y, mny = p[0].y;
#pragma unroll
    for (int j = 1; j < 4; ++j) {
        mxx = fmaxf(mxx, p[j].x); mnx = fminf(mnx, p[j].x);
        mxy = fmaxf(mxy, p[j].y); mny = fminf(mny, p[j].y);
    }
#pragma unroll
    for (int j = 0; j < 4; ++j) {
        mxx = fmaxf(mxx, t[j].x); mnx = fminf(mnx, t[j].x);
        mxy = fmaxf(mxy, t[j].y); mny = fminf(mny, t[j].y);
    }
    float c2 = (mxx - mnx) * (mxx - mnx) + (mxy - mny) * (mxy - mny) + EPSF;

    return 1.0f - iou + d2 / c2;
}

__global__ void diou_partial_kernel(const float* __restrict__ pred,
                                    const float* __restrict__ tgt,
                                    float* __restrict__ partial, int N) {
    const int tid = threadIdx.x;
    int gid = blockIdx.x * blockDim.x + tid;
    const int stride = gridDim.x * blockDim.x;
    const float4* pv = reinterpret_cast<const float4*>(pred);
    const float4* tv = reinterpret_cast<const float4*>(tgt);
    float local = 0.0f;
    for (int i = gid; i < N; i += stride) local += diou_single(pv, tv, i);
    __shared__ float s[TPB];
    s[tid] = local;
    __syncthreads();
    for (int off = TPB / 2; off > 0; off >>= 1) {
        if (tid < off) s[tid] += s[tid + off];
        __syncthreads();
    }
    if (tid == 0) partial[blockIdx.x] = s[0];
}

typedef __attribute__((ext_vector_type(2))) float v2f;
typedef __attribute__((ext_vector_type(8))) float v8f;

// Single-wave finisher: deterministically reduces M block partials.
// 64 per-slot accumulators (2 VGPRs x 32 lanes) are summed by
// V_WMMA_F32_16X16X4_F32 with A = ones: D[m][n] = sum_k B[k][n]; the 16 column
// sums partition all 64 B elements regardless of the (lane,vgpr)->(k,n)
// mapping, so summing row 0 of D gives the exact total in full f32 precision.
__global__ void diou_final_kernel(const float* __restrict__ partial,
                                  float* __restrict__ out, float scale, int M) {
    const int lane = threadIdx.x;  // 0..31, one wave32
    float s0 = 0.0f, s1 = 0.0f;
    for (int idx = lane; idx < M; idx += 64)      s0 += partial[idx];
    for (int idx = lane + 32; idx < M; idx += 64) s1 += partial[idx];
#if __has_builtin(__builtin_amdgcn_wmma_f32_16x16x4_f32)
    v2f a; a.x = 1.0f; a.y = 1.0f;          // A-matrix: all ones
    v2f b; b.x = s0;   b.y = s1;            // B-matrix: the 64 slot sums
    v8f c = {};                              // C = 0
    c = __builtin_amdgcn_wmma_f32_16x16x4_f32(
        /*neg_a=*/false, a, /*neg_b=*/false, b,
        /*c_mod=*/(short)0, c, /*reuse_a=*/false, /*reuse_b=*/false);
    float col = c[0];  // VGPR0: lanes 0-15 -> D[0][n], lanes 16-31 -> D[8][n] (dups)
    col += __shfl_xor(col, 1, 32);
    col += __shfl_xor(col, 2, 32);
    col += __shfl_xor(col, 4, 32);
    col += __shfl_xor(col, 8, 32);           // lane 0: sum of the 16 column sums
    float total = col;
#else
    float v = s0 + s1;
    v += __shfl_xor(v, 1, 32);
    v += __shfl_xor(v, 2, 32);
    v += __shfl_xor(v, 4, 32);
    v += __shfl_xor(v, 8, 32);
    v += __shfl_xor(v, 16, 32);
    float total = v;
#endif
    if (lane == 0) out[0] = total * scale;
}

extern "C" void kernel_launch(void* const* d_in, const int* in_sizes, int n_in,
                              void* d_out, int out_size, void* d_ws, size_t ws_size,
                              hipStream_t stream) {
    const float* pred = (const float*)d_in[0];
    const float* tgt  = (const float*)d_in[1];
    float* out = (float*)d_out;
    float* partial = (float*)d_ws;

    const int N = in_sizes[0] / 8;           // rows of 4 (x,y) points
    int nblk = NBLK;
    if (ws_size < (size_t)nblk * sizeof(float)) {
        nblk = (int)(ws_size / sizeof(float));
        if (nblk < 64) nblk = 64;            // defensive; ws is spec'd preallocated
    }
    diou_partial_kernel<<<nblk, TPB, 0, stream>>>(pred, tgt, partial, N);

    const float scale = (float)(1.0 / (double)N);  // LOSS_WEIGHT * mean
    diou_final_kernel<<<1, 32, 0, stream>>>(partial, out, scale, nblk);
}